// STGCN_75350906241135
// MI455X (gfx1250) — compile-verified
//
#include <hip/hip_runtime.h>
#include <hip/hip_bf16.h>

#define Bc   4
#define Tc   12
#define CINc 2
#define Hc   64

typedef __attribute__((ext_vector_type(2))) float v2f;
typedef __attribute__((ext_vector_type(8))) float v8f;

// ---------------------------------------------------------------------------
// Temporal conv over node axis, kernel 3, pad (1,0), + bias + ReLU.
// h0[row*64 + c], row = (b*T + t)*Nout + n
// ---------------------------------------------------------------------------
__global__ void k_tconv(const float* __restrict__ x, const float* __restrict__ cw,
                        const float* __restrict__ cb, float* __restrict__ h0,
                        int N, int Nout, int ntot) {
    int gid = blockIdx.x * blockDim.x + threadIdx.x;
    if (gid >= ntot * Hc) return;
    int c   = gid & 63;
    int row = gid >> 6;
    int n   = row % Nout;
    int bt  = row / Nout;
    const float* xb = x + (size_t)bt * N * CINc;
    float acc = cb[c];
#pragma unroll
    for (int k = 0; k < 3; ++k) {
        int nn = n + k - 1;
        if (nn < 0 || nn >= N) continue;
        float x0 = xb[nn * 2 + 0];
        float x1 = xb[nn * 2 + 1];
        acc += x0 * cw[(c * 2 + 0) * 3 + k] + x1 * cw[(c * 2 + 1) * 3 + k];
    }
    h0[gid] = fmaxf(acc, 0.0f);
}

// ---------------------------------------------------------------------------
// Degree: fill 1.0 (self loop), atomic-add edge weights at dst.
// ---------------------------------------------------------------------------
__global__ void k_fill(float* __restrict__ p, float v, int n) {
    int i = blockIdx.x * blockDim.x + threadIdx.x;
    if (i < n) p[i] = v;
}

__global__ void k_deg(const int* __restrict__ ei, const float* __restrict__ ew,
                      float* __restrict__ deg, int E, int ntot) {
    int e = blockIdx.x * blockDim.x + threadIdx.x;
    if (e >= E) return;
    int dst = ei[E + e];
    if ((unsigned)dst < (unsigned)ntot) unsafeAtomicAdd(&deg[dst], ew[e]);
}

// deg (in dinv_io) -> dinv (in place) and invdeg
__global__ void k_dinv(float* __restrict__ dinv_io, float* __restrict__ invdeg, int ntot) {
    int i = blockIdx.x * blockDim.x + threadIdx.x;
    if (i >= ntot) return;
    float d = dinv_io[i];
    dinv_io[i] = rsqrtf(d);
    invdeg[i]  = 1.0f / d;
}

// norm[e] = dinv[clamp(src)] * w * dinv[clamp(dst)]   (JAX clamp-gather semantics)
__global__ void k_norm(const int* __restrict__ ei, const float* __restrict__ ew,
                       const float* __restrict__ dinv, float* __restrict__ norm,
                       int E, int ntot) {
    int e = blockIdx.x * blockDim.x + threadIdx.x;
    if (e >= E) return;
    int s = min(ei[e],     ntot - 1);
    int d = min(ei[E + e], ntot - 1);
    norm[e] = dinv[s] * ew[e] * dinv[d];
}

// ---------------------------------------------------------------------------
// GEMM: Z = act(A) @ W  (M x 64 @ 64 x 64), f32 WMMA 16x16x4.
// Epilogue fuses the GCN self-loop init:  AGG = Z*invdeg[row] + bias[col].
// Block = 128 threads = 4 waves; block owns a 16-row stripe; wave w owns the
// 16-col tile n0 = 16*w. W staged through LDS. No divergence around WMMA.
// ---------------------------------------------------------------------------
template <bool RELU>
__global__ __launch_bounds__(128)
void k_gemm64(const float* __restrict__ A, const float* __restrict__ W,
              const float* __restrict__ invdeg, const float* __restrict__ bias,
              float* __restrict__ Z, float* __restrict__ AGG, int ntot) {
    __shared__ __align__(16) float sW[64 * 64];
    int tid = threadIdx.x;
    for (int i = tid * 4; i < 64 * 64; i += 128 * 4)
        *(float4*)&sW[i] = *(const float4*)&W[i];
    __syncthreads();

    int wave = tid >> 5;
    int lane = tid & 31;
    int hh   = lane >> 4;      // which half-wave (selects K pair / M+8)
    int l16  = lane & 15;
    int m0   = blockIdx.x * 16;
    int n0   = wave * 16;

    int arow = min(m0 + l16, ntot - 1);
    const float* Arow = A + (size_t)arow * 64;

    v8f c = {};
#pragma unroll
    for (int k0 = 0; k0 < 64; k0 += 4) {
        // A 16x4: VGPR0 = {K=k0 | K=k0+2}, VGPR1 = {K=k0+1 | K=k0+3}
        float a0 = Arow[k0 + 2 * hh + 0];
        float a1 = Arow[k0 + 2 * hh + 1];
        if (RELU) { a0 = fmaxf(a0, 0.0f); a1 = fmaxf(a1, 0.0f); }
        v2f a; a.x = a0; a.y = a1;
        // B 4x16: same K striping, N = n0 + l16 across lanes
        v2f b;
        b.x = sW[(k0 + 2 * hh + 0) * 64 + n0 + l16];
        b.y = sW[(k0 + 2 * hh + 1) * 64 + n0 + l16];
        c = __builtin_amdgcn_wmma_f32_16x16x4_f32(false, a, false, b,
                                                  (short)0, c, false, false);
    }

    // D layout: VGPR r -> M = m0 + r + 8*hh, N = n0 + l16
#pragma unroll
    for (int r = 0; r < 8; ++r) {
        int row = m0 + r + 8 * hh;
        if (row >= ntot) continue;
        int col = n0 + l16;
        float z = c[r];
        size_t idx = (size_t)row * 64 + col;
        Z[idx]   = z;
        AGG[idx] = z * invdeg[row] + bias[col];
    }
}

// ---------------------------------------------------------------------------
// Edge scatter: AGG[dst][c] += norm[e] * Z[clamp(src)][c]; 64 threads/edge,
// coalesced 128B gather + 128B atomic stream per wave; all L2-resident.
// ---------------------------------------------------------------------------
__global__ void k_scatter(const int* __restrict__ ei, const float* __restrict__ norm,
                          const float* __restrict__ Z, float* __restrict__ AGG,
                          int E, int ntot) {
    int gid = blockIdx.x * blockDim.x + threadIdx.x;   // E*64 < 2^31
    if (gid >= E * 64) return;
    int e = gid >> 6;
    int c = gid & 63;
    int dst = ei[E + e];
    if ((unsigned)dst >= (unsigned)ntot) return;       // segment_sum drops OOB
    int s = min(ei[e], ntot - 1);                      // gather clamps
    float w = norm[e];
    unsafeAtomicAdd(&AGG[(size_t)dst * 64 + c], w * Z[(size_t)s * 64 + c]);
}

// ---------------------------------------------------------------------------
// Final FC on the t = T-1 slice: out[b*Nout+n] = dot(h2[row], fc_w) + fc_b
// ---------------------------------------------------------------------------
__global__ void k_fc(const float* __restrict__ H2, const float* __restrict__ fcw,
                     const float* __restrict__ fcb, float* __restrict__ out,
                     int Nout, int total) {
    int i = blockIdx.x * blockDim.x + threadIdx.x;
    if (i >= total) return;
    int b = i / Nout, n = i % Nout;
    const float* h = H2 + (size_t)((b * Tc + (Tc - 1)) * Nout + n) * 64;
    float acc = fcb[0];
#pragma unroll
    for (int c = 0; c < 64; c += 4) {
        float4 hv = *(const float4*)&h[c];
        float4 wv = *(const float4*)&fcw[c];
        acc += hv.x * wv.x + hv.y * wv.y + hv.z * wv.z + hv.w * wv.w;
    }
    out[i] = acc;
}

// ---------------------------------------------------------------------------
extern "C" void kernel_launch(void* const* d_in, const int* in_sizes, int n_in,
                              void* d_out, int out_size, void* d_ws, size_t ws_size,
                              hipStream_t stream) {
    const float* x      = (const float*)d_in[0];
    const int*   ei     = (const int*)  d_in[1];
    const float* ew     = (const float*)d_in[2];
    const float* conv_w = (const float*)d_in[3];
    const float* conv_b = (const float*)d_in[4];
    const float* W1     = (const float*)d_in[5];
    const float* b1     = (const float*)d_in[6];
    const float* W2     = (const float*)d_in[7];
    const float* b2     = (const float*)d_in[8];
    const float* fc_w   = (const float*)d_in[9];
    const float* fc_b   = (const float*)d_in[10];
    float* out = (float*)d_out;

    const int N    = in_sizes[0] / (Bc * Tc * CINc);   // 2000
    const int E    = in_sizes[2];                      // 1,536,000
    const int Nout = out_size / Bc;                    // conv output nodes (1999)
    const int ntot = Bc * Tc * Nout;                   // rows (multiple of 16)

    float* ws = (float*)d_ws;
    size_t o = 0;
    float* dinv   = ws + o; o += (size_t)ntot;
    float* invdeg = ws + o; o += (size_t)ntot;
    float* norm   = ws + o; o += (size_t)E;
    o = (o + 63) & ~(size_t)63;
    float* h0   = ws + o; o += (size_t)ntot * 64;      // conv out / layer-2 agg
    float* zbuf = ws + o; o += (size_t)ntot * 64;      // GEMM output (per layer)
    float* agg1 = ws + o; o += (size_t)ntot * 64;      // layer-1 agg (h1)

    const int TPB = 256;
    // 1) temporal conv + ReLU
    k_tconv<<<(ntot * 64 + TPB - 1) / TPB, TPB, 0, stream>>>(x, conv_w, conv_b, h0,
                                                             N, Nout, ntot);
    // 2) degree (self-loop 1.0 + scatter of edge weights)
    k_fill<<<(ntot + TPB - 1) / TPB, TPB, 0, stream>>>(dinv, 1.0f, ntot);
    k_deg<<<(E + TPB - 1) / TPB, TPB, 0, stream>>>(ei, ew, dinv, E, ntot);
    k_dinv<<<(ntot + TPB - 1) / TPB, TPB, 0, stream>>>(dinv, invdeg, ntot);
    k_norm<<<(E + TPB - 1) / TPB, TPB, 0, stream>>>(ei, ew, dinv, norm, E, ntot);

    const int gemmBlocks = (ntot + 15) / 16;
    const int scatBlocks = (E * 64 + TPB - 1) / TPB;

    // 3) layer 1: z = h0 @ W1 ; agg1 = z*invdeg + b1 ; scatter edges
    k_gemm64<false><<<gemmBlocks, 128, 0, stream>>>(h0, W1, invdeg, b1, zbuf, agg1, ntot);
    k_scatter<<<scatBlocks, TPB, 0, stream>>>(ei, norm, zbuf, agg1, E, ntot);

    // 4) layer 2: z = relu(agg1) @ W2 ; h2 (into h0) = z*invdeg + b2 ; scatter
    k_gemm64<true><<<gemmBlocks, 128, 0, stream>>>(agg1, W2, invdeg, b2, zbuf, h0, ntot);
    k_scatter<<<scatBlocks, TPB, 0, stream>>>(ei, norm, zbuf, h0, E, ntot);

    // 5) final FC on last time slice
    k_fc<<<(Bc * Nout + TPB - 1) / TPB, TPB, 0, stream>>>(h0, fc_w, fc_b, out,
                                                          Nout, Bc * Nout);
}